// MLinear_62706522522340
// MI455X (gfx1250) — compile-verified
//
#include <hip/hip_runtime.h>

typedef float v2f __attribute__((ext_vector_type(2)));
typedef float v8f __attribute__((ext_vector_type(8)));

#define FIN        64
#define FOUT       64
#define NSLICE     256
#define BATCH      4096
#define ROWS_PB    128          // rows of x per block (8 waves * 16 rows)
#define THREADS    256
#define LDS_STRIDE 68           // padded W row stride (floats): 16B-aligned, bank-conflict-free

__global__ __launch_bounds__(THREADS)
void grouped_linear_wmma(const float* __restrict__ x,
                         const float* __restrict__ W,
                         const float* __restrict__ bias,
                         float* __restrict__ out)
{
    __shared__ float sW[FOUT * LDS_STRIDE];   // 64 x 68 floats = 17 KB

    const int n       = blockIdx.y;
    const int tileRow = blockIdx.x * ROWS_PB;
    const int t       = threadIdx.x;

    // ---- cooperative load of W[n] (64x64 f32, [o][f] row-major) into padded LDS ----
    const float* Wn = W + (size_t)n * (FOUT * FIN);
    #pragma unroll
    for (int i = 0; i < 4; ++i) {
        int idx4 = i * THREADS + t;            // 1024 float4's total
        int g    = idx4 * 4;                   // flat float index
        int o    = g >> 6;
        int f    = g & 63;
        float4 v = *(const float4*)(Wn + g);
        *(float4*)(&sW[o * LDS_STRIDE + f]) = v;
    }
    __syncthreads();

    const int lane = t & 31;
    const int wave = t >> 5;
    const int hi   = lane >> 4;      // 0: lanes 0-15, 1: lanes 16-31
    const int ol   = lane & 15;      // M index (A) / N index (B, C)

    const int rowBase = tileRow + wave * 16;

    // ---- A fragments: 16 k-steps of the 16x4 f32 layout ----
    // lane(L): M = L%16, VGPR pair covers K = s*4 + (L>=16 ? 2 : 0) + {0,1}
    const float* arow = x + ((size_t)(rowBase + ol) * NSLICE + n) * FIN + hi * 2;
    v2f a[16];
    #pragma unroll
    for (int s = 0; s < 16; ++s)
        a[s] = *(const v2f*)(arow + s * 4);

    v8f acc[4] = { {}, {}, {}, {} };

    // ---- K loop: D = A x B + C via v_wmma_f32_16x16x4_f32 ----
    #pragma unroll
    for (int s = 0; s < 16; ++s) {
        #pragma unroll
        for (int nt = 0; nt < 4; ++nt) {
            // B fragment: lane(L) holds column N = nt*16 + L%16, K = s*4 + hi*2 + {0,1}
            // B[k][o] = W[n][o][k]  (torch Linear stores [out, in])
            v2f bfrag = *(const v2f*)(&sW[(nt * 16 + ol) * LDS_STRIDE + s * 4 + hi * 2]);
            acc[nt] = __builtin_amdgcn_wmma_f32_16x16x4_f32(
                /*neg_a=*/false, a[s],
                /*neg_b=*/false, bfrag,
                /*c_mod=*/(short)0, acc[nt],
                /*reuse_a=*/false, /*reuse_b=*/false);
        }
    }

    // ---- bias add + store ----
    // C/D layout: VGPR j, lanes 0-15 -> M=j, N=lane; lanes 16-31 -> M=j+8, N=lane-16
    const float* bn = bias + n * FOUT;
    #pragma unroll
    for (int nt = 0; nt < 4; ++nt) {
        const float bv = bn[nt * 16 + ol];
        acc[nt] = acc[nt] + bv;                 // splat add across all 8 M-rows
        #pragma unroll
        for (int j = 0; j < 8; ++j) {
            int m = rowBase + j + 8 * hi;
            out[((size_t)m * NSLICE + n) * FOUT + nt * 16 + ol] = acc[nt][j];
        }
    }
}

extern "C" void kernel_launch(void* const* d_in, const int* in_sizes, int n_in,
                              void* d_out, int out_size, void* d_ws, size_t ws_size,
                              hipStream_t stream) {
    const float* x   = (const float*)d_in[0];   // [4096, 256, 64]
    const float* W   = (const float*)d_in[1];   // [256, 64, 64]
    const float* b   = (const float*)d_in[2];   // [256, 64]
    float*       out = (float*)d_out;           // [4096, 256, 64]

    dim3 grid(BATCH / ROWS_PB, NSLICE);
    grouped_linear_wmma<<<grid, THREADS, 0, stream>>>(x, W, b, out);
}